// ExpansionContrastModule_7352984011499
// MI455X (gfx1250) — compile-verified
//
#include <hip/hip_runtime.h>
#include <hip/hip_bf16.h>

typedef __attribute__((ext_vector_type(2))) float v2f;
typedef __attribute__((ext_vector_type(8))) float v8f;

#define IMG_W   256
#define IMG_H   256
#define HW      65536           // 256*256
#define NB      4               // batch

// workspace layout (float offsets)
#define WEXT_OFF   0                         // 16*64 folded input weights (row 8 = sen fold)
#define UCOEF_OFF  1024                      // 8*8*4  u[g][i][f]
#define BIAS_OFF   1280                      // 1
#define XBUF_OFF   2048                      // NB*8*HW   x channels, channel-major
#define SEN_OFF    (2048 + NB*8*HW)          // NB*HW     folded sen dot

// ---------------------------------------------------------------------------
// Kernel 1: fold all the 1x1 convs into Wext / u / bias (tiny, one block).
// ---------------------------------------------------------------------------
__global__ void ecm_setup(const float* __restrict__ W_in,
                          const float* __restrict__ W_lo,
                          const float* __restrict__ b_lo,
                          const float* __restrict__ W_l1,
                          const float* __restrict__ b_l1,
                          const float* __restrict__ W_l2,
                          const float* __restrict__ b_l2,
                          const float* __restrict__ W_out,
                          const float* __restrict__ b_out,
                          float* __restrict__ ws) {
    __shared__ float v[256];                 // v[i][idx], i<8 stencils, idx<32
    int tid = threadIdx.x;                   // 256 threads

    // v[i][idx] = sum_o2 W_out[i*8+o2] * W_l2[i,o2,idx]
    {
        int i = tid >> 5, idx = tid & 31;
        float s = 0.f;
        for (int o2 = 0; o2 < 8; ++o2)
            s += W_out[i * 8 + o2] * W_l2[(i * 8 + o2) * 32 + idx];
        v[i * 32 + idx] = s;
    }
    __syncthreads();

    // u[g][i][f] = sum_o v[i][4g+o] * W_l1[g,o,f]
    {
        int g = tid >> 5, rem = tid & 31;
        int i = rem >> 2, f = rem & 3;
        float s = 0.f;
        for (int o = 0; o < 4; ++o)
            s += v[i * 32 + 4 * g + o] * W_l1[(g * 4 + o) * 4 + f];
        ws[UCOEF_OFF + (g * 8 + i) * 4 + f] = s;
    }

    // Wext: rows 0..7 = W_in, row 8 = W_out[64:] . W_lo (sen fold), rows 9..15 = 0
    for (int idx = tid; idx < 1024; idx += 256) {
        int m = idx >> 6, c = idx & 63;
        float val = 0.f;
        if (m < 8) {
            val = W_in[m * 64 + c];
        } else if (m == 8) {
            for (int o = 0; o < 64; ++o) val += W_out[64 + o] * W_lo[o * 64 + c];
        }
        ws[WEXT_OFF + idx] = val;
    }

    if (tid == 0) {
        float s = b_out[0];
        for (int o = 0; o < 64; ++o) s += W_out[64 + o] * b_lo[o];     // sen bias
        for (int c = 0; c < 64; ++c) s += W_out[c] * b_l2[c];          // layer2 bias
        for (int i = 0; i < 8; ++i)                                     // layer1 bias
            for (int g = 0; g < 8; ++g)
                for (int o = 0; o < 4; ++o)
                    s += v[i * 32 + 4 * g + o] * b_l1[g * 4 + o];
        ws[BIAS_OFF] = s;
    }
}

// ---------------------------------------------------------------------------
// Kernel 2: x/sen = Wext(16x64) . cen(64 x 16-pixel tile) via chained
// V_WMMA_F32_16X16X4_F32 (exact fp32). One wave per 16 consecutive pixels.
// D rows 0..7 (lanes 0-15) = x channels; D row 8 (VGPR0, lanes 16-31) = sen.
// ---------------------------------------------------------------------------
__global__ void ecm_pass1(const float* __restrict__ cen,
                          const float* __restrict__ b_in,
                          float* __restrict__ ws) {
    int lane = threadIdx.x & 31;
    int wave = threadIdx.x >> 5;
    int group = blockIdx.x * 8 + wave;          // 16 pixels per group
    size_t pix = (size_t)group * 16;
    int b  = (int)(pix >> 16);                  // / HW
    int hw = (int)(pix & (HW - 1));

    int m    = lane & 15;                       // matrix row (A: M, B: N/pixel)
    int half = lane >> 4;                       // K sub-pair select

    const float* wext  = ws + WEXT_OFF;
    const float* cbase = cen + (size_t)b * 64 * HW + hw + m;

    v8f acc = {};
#pragma unroll
    for (int kk = 0; kk < 16; ++kk) {
        int k0 = kk * 4 + half * 2;
        // A: Wext[m][k0], Wext[m][k0+1]   (16x4 f32 A layout)
        v2f a;
        a[0] = wext[m * 64 + k0];
        a[1] = wext[m * 64 + k0 + 1];
        // B: cen[k0][pixel m], cen[k0+1][pixel m]   (4x16 f32 B layout)
        v2f bb;
        bb[0] = cbase[(size_t)k0 * HW];
        bb[1] = cbase[(size_t)(k0 + 1) * HW];
        if (kk < 12)  // prefetch B operand 4 chunks (16 channels) ahead
            __builtin_prefetch(&cbase[(size_t)(k0 + 16) * HW], 0, 0);
        acc = __builtin_amdgcn_wmma_f32_16x16x4_f32(
            false, a, false, bb, (short)0, acc, false, false);
    }

    if (lane < 16) {
        float* xb = ws + XBUF_OFF + (size_t)b * 8 * HW + hw + lane;
#pragma unroll
        for (int r = 0; r < 8; ++r)
            xb[(size_t)r * HW] = acc[r] + b_in[r];   // bias before zero-pad, as ref
    } else {
        ws[SEN_OFF + (size_t)b * HW + hw + (lane - 16)] = acc[0];
    }
}

// ---------------------------------------------------------------------------
// Kernel 3: stencil + folded quadratic form. 16x16 output tile per block,
// x tile with 1-halo staged in LDS (zero-padded = reference _neigh).
// ---------------------------------------------------------------------------
__global__ void ecm_pass2(const float* __restrict__ ws,
                          float* __restrict__ out) {
    __shared__ float tile[8 * 18 * 18];     // [ch][ly][lx]

    int b   = blockIdx.z;
    int tx0 = blockIdx.x * 16, ty0 = blockIdx.y * 16;
    int tid = threadIdx.y * 16 + threadIdx.x;

    const float* xb = ws + XBUF_OFF + (size_t)b * 8 * HW;
    for (int idx = tid; idx < 8 * 324; idx += 256) {
        int ch  = idx / 324;
        int pos = idx - ch * 324;
        int ly  = pos / 18, lx = pos - ly * 18;
        int gy  = ty0 + ly - 1, gx = tx0 + lx - 1;
        float v = 0.f;
        if (gy >= 0 && gy < IMG_H && gx >= 0 && gx < IMG_W)
            v = xb[(size_t)ch * HW + gy * IMG_W + gx];
        tile[idx] = v;
    }
    __syncthreads();

    const int gy = ty0 + threadIdx.y;
    const int gx = tx0 + threadIdx.x;
    const int cy = threadIdx.y + 1, cx = threadIdx.x + 1;

    const float* uc = ws + UCOEF_OFF;
    float z = ws[BIAS_OFF] + ws[SEN_OFF + (size_t)b * HW + gy * IMG_W + gx];

    const int dyo[8] = {-1, -1, -1, 0, 1, 1, 1, 0};
    const int dxo[8] = {-1, 0, 1, 1, 1, 0, -1, -1};

#pragma unroll
    for (int g = 0; g < 8; ++g) {
        const float* t = tile + g * 324;
        float xc = t[cy * 18 + cx];
        float d[8];
#pragma unroll
        for (int i = 0; i < 8; ++i)
            d[i] = xc - t[(cy + dyo[i]) * 18 + (cx + dxo[i])];
#pragma unroll
        for (int i = 0; i < 8; ++i) {
            const float* u = uc + (g * 8 + i) * 4;   // uniform -> scalar loads
            float t0 = u[0] * (d[(i + 1) & 7] + d[(i + 7) & 7]);
            float t1 = u[1] * (d[(i + 2) & 7] + d[(i + 6) & 7]);
            float t2 = u[2] * (d[(i + 3) & 7] + d[(i + 5) & 7]);
            float t3 = u[3] * 2.0f * (d[(i + 4) & 7] - d[i]);
            z += (t0 + t1 + t2 + t3) * d[i];
        }
    }

    out[(size_t)b * HW + gy * IMG_W + gx] = z;
}

// ---------------------------------------------------------------------------
extern "C" void kernel_launch(void* const* d_in, const int* in_sizes, int n_in,
                              void* d_out, int out_size, void* d_ws, size_t ws_size,
                              hipStream_t stream) {
    const float* cen   = (const float*)d_in[0];
    const float* W_in  = (const float*)d_in[1];
    const float* b_in  = (const float*)d_in[2];
    const float* W_lo  = (const float*)d_in[3];
    const float* b_lo  = (const float*)d_in[4];
    const float* W_l1  = (const float*)d_in[5];
    const float* b_l1  = (const float*)d_in[6];
    const float* W_l2  = (const float*)d_in[7];
    const float* b_l2  = (const float*)d_in[8];
    const float* W_out = (const float*)d_in[9];
    const float* b_out = (const float*)d_in[10];
    float* ws  = (float*)d_ws;
    float* out = (float*)d_out;

    ecm_setup<<<1, 256, 0, stream>>>(W_in, W_lo, b_lo, W_l1, b_l1,
                                     W_l2, b_l2, W_out, b_out, ws);

    // NB*HW/16 pixel-groups, 8 waves (groups) per 256-thread block
    int groups = NB * HW / 16;
    ecm_pass1<<<groups / 8, 256, 0, stream>>>(cen, b_in, ws);

    dim3 g2(IMG_W / 16, IMG_H / 16, NB);
    dim3 b2(16, 16);
    ecm_pass2<<<g2, b2, 0, stream>>>(ws, out);
}